// OriginalSpatialAttention_56332791054808
// MI455X (gfx1250) — compile-verified
//
#include <hip/hip_runtime.h>
#include <hip/hip_bf16.h>
#include <math.h>

// Problem constants (from reference): x[8,64,32,64,64] f32, W[1,2,3,3,3] f32.
constexpr int B_ = 8, C_ = 64, D_ = 32, H_ = 64, W_ = 64;
constexpr int S_  = D_ * H_ * W_;     // 131072 spatial positions per (b,c)
constexpr int CS_ = C_ * S_;          // elements per batch of x

typedef __attribute__((ext_vector_type(2))) float v2f;
typedef __attribute__((ext_vector_type(8))) float v8f;

// ---------------------------------------------------------------------------
// Pass 1: channel mean (via V_WMMA_F32_16X16X4_F32) + channel max (VALU),
// streaming 256 MB at HBM rate. One wave handles 16 consecutive spatial
// positions; loads are issued directly in the WMMA B-matrix layout:
//   B (4x16 f32): VGPR0 = lanes0-15 (K=0,N) | lanes16-31 (K=2,N)
//                 VGPR1 = lanes0-15 (K=1,N) | lanes16-31 (K=3,N)
// A is the constant 1/64 broadcast, so D[m,n] accumulates mean(x[:,n]).
// Results stored as f16 planes (avg, max) in workspace (4 MB, L2-resident).
// ---------------------------------------------------------------------------
__global__ __launch_bounds__(256) void reduce_mean_max_wmma(
    const float* __restrict__ x,
    _Float16* __restrict__ avgp,
    _Float16* __restrict__ maxp)
{
    const int lane = threadIdx.x & 31;
    const int wv   = (blockIdx.x << 3) + (threadIdx.x >> 5);  // global wave id
    const int p0   = wv << 4;                                 // 16 positions/wave
    const int b    = p0 / S_;
    const int sp   = p0 - b * S_;

    const int nn   = lane & 15;     // N (position within the 16-wide tile)
    const int half = lane >> 4;     // 0 -> K={0,1}, 1 -> K={2,3} within a chunk

    const float* base = x + (size_t)b * CS_ + sp;
    const size_t off0 = (size_t)(half * 2) * S_ + (size_t)nn;  // channel c0+2*half
    const size_t off1 = off0 + (size_t)S_;                     // channel c0+2*half+1

    v2f a;  a[0] = 0.015625f; a[1] = 0.015625f;   // A = 1/64 everywhere
    v8f acc0 = {};
    v8f acc1 = {};
    float mx0 = -INFINITY, mx1 = -INFINITY;

    #pragma unroll
    for (int c = 0; c < 16; ++c) {                // 16 chunks x 4 channels = 64
        const float* p = base + (size_t)c * 4u * (size_t)S_;
        float v0 = p[off0];
        float v1 = p[off1];
        mx0 = fmaxf(mx0, v0);
        mx1 = fmaxf(mx1, v1);
        v2f bb; bb[0] = v0; bb[1] = v1;
        if (c & 1)
            acc1 = __builtin_amdgcn_wmma_f32_16x16x4_f32(
                false, a, false, bb, (short)0, acc1, false, false);
        else
            acc0 = __builtin_amdgcn_wmma_f32_16x16x4_f32(
                false, a, false, bb, (short)0, acc0, false, false);
    }

    // D rows are identical (A rows replicated); element 0 of the C/D layout is
    // (M=0,N=lane) for lanes 0-15 and (M=8,N=lane-16) for lanes 16-31 -> same
    // mean value for position nn in both halves.
    float mean = acc0[0] + acc1[0];

    // Max: mx0/mx1 cover channel classes split across lane halves; combine
    // within lane, then across halves (positions repeat with xor-16 lanes).
    float m = fmaxf(mx0, mx1);
    m = fmaxf(m, __shfl_xor(m, 16, 32));

    if (lane < 16) {
        const int idx = b * S_ + sp + nn;
        avgp[idx] = (_Float16)mean;
        maxp[idx] = (_Float16)m;
    }
}

// ---------------------------------------------------------------------------
// Pass 2: 3x3x3 conv over the two f16 planes (SAME, zero pad) + sigmoid.
// Output channel count is 1 so im2col-WMMA would be >=16x redundant; a
// register-blocked VALU stencil is the right tool (reads 8 MB from L2,
// ~54 FMA/output, writes 4 MB). One thread computes 4 consecutive w outputs.
// ---------------------------------------------------------------------------
__global__ __launch_bounds__(256) void conv3d_sigmoid(
    const _Float16* __restrict__ avgp,
    const _Float16* __restrict__ maxp,
    const float* __restrict__ Wt,      // 54 weights: [ci][kd][kh][kw]
    float* __restrict__ out)
{
    const int t  = blockIdx.x * 256 + threadIdx.x;   // 0 .. 262143
    const int w0 = (t & 15) * 4;
    const int h  = (t >> 4) & 63;
    const int d  = (t >> 10) & 31;
    const int b  = t >> 15;

    // Uniform weight loads -> scalar cache.
    float wk[54];
    #pragma unroll
    for (int i = 0; i < 54; ++i) wk[i] = Wt[i];

    float acc0 = 0.f, acc1 = 0.f, acc2 = 0.f, acc3 = 0.f;

    #pragma unroll
    for (int ci = 0; ci < 2; ++ci) {
        const _Float16* pl = (ci == 0) ? avgp : maxp;
        const _Float16* pb = pl + (size_t)b * S_;
        #pragma unroll
        for (int kd = 0; kd < 3; ++kd) {
            const int dd  = d + kd - 1;
            const bool okd = (unsigned)dd < (unsigned)D_;
            const int ddc = okd ? dd : 0;
            #pragma unroll
            for (int kh = 0; kh < 3; ++kh) {
                const int hh  = h + kh - 1;
                const bool okh = (unsigned)hh < (unsigned)H_;
                const int hhc = okh ? hh : 0;
                const _Float16* row = pb + ((size_t)ddc * H_ + hhc) * W_;
                const bool okdh = okd && okh;

                // Need w in [w0-1, w0+4]: 6 taps with zero padding.
                float v[6];
                #pragma unroll
                for (int j = 0; j < 6; ++j) {
                    const int ww  = w0 - 1 + j;
                    const bool ok = okdh && ((unsigned)ww < (unsigned)W_);
                    const int wwc = ((unsigned)ww < (unsigned)W_) ? ww : 0;
                    v[j] = ok ? (float)row[wwc] : 0.f;
                }

                const int wb = ci * 27 + kd * 9 + kh * 3;
                #pragma unroll
                for (int kw = 0; kw < 3; ++kw) {
                    const float wv = wk[wb + kw];
                    acc0 = fmaf(v[0 + kw], wv, acc0);
                    acc1 = fmaf(v[1 + kw], wv, acc1);
                    acc2 = fmaf(v[2 + kw], wv, acc2);
                    acc3 = fmaf(v[3 + kw], wv, acc3);
                }
            }
        }
    }

    float4 r;
    r.x = 1.f / (1.f + __expf(-acc0));
    r.y = 1.f / (1.f + __expf(-acc1));
    r.z = 1.f / (1.f + __expf(-acc2));
    r.w = 1.f / (1.f + __expf(-acc3));

    const size_t oidx = (((size_t)b * D_ + d) * H_ + h) * W_ + w0;  // 16B aligned
    *(float4*)(out + oidx) = r;
}

// ---------------------------------------------------------------------------
// Launch: pass1 (65536 waves = 8192 blocks x 256) then pass2 (1024 blocks).
// Workspace use: 2 x 1,048,576 f16 = 4 MB (avg plane then max plane).
// ---------------------------------------------------------------------------
extern "C" void kernel_launch(void* const* d_in, const int* in_sizes, int n_in,
                              void* d_out, int out_size, void* d_ws, size_t ws_size,
                              hipStream_t stream) {
    (void)in_sizes; (void)n_in; (void)out_size; (void)ws_size;
    const float* x  = (const float*)d_in[0];
    const float* Wt = (const float*)d_in[1];
    float* out = (float*)d_out;

    _Float16* avgp = (_Float16*)d_ws;
    _Float16* maxp = avgp + (size_t)B_ * S_;

    reduce_mean_max_wmma<<<8192, 256, 0, stream>>>(x, avgp, maxp);
    conv3d_sigmoid<<<1024, 256, 0, stream>>>(avgp, maxp, Wt, out);
}